// DisplacementNet_14972255993924
// MI455X (gfx1250) — compile-verified
//
#include <hip/hip_runtime.h>

typedef float v2f __attribute__((ext_vector_type(2)));
typedef float v8f __attribute__((ext_vector_type(8)));

#define KSEL 12          // k nearest neighbors kept
#define ROWS 128         // rows per block (== threads per block, 4 waves)
#define CT   64          // columns per chunk
#define LDSW 68          // padded LDS row stride in words (16B-aligned rows, conflict-free stores)

__global__ __launch_bounds__(ROWS)
void knn_wmma_kernel(const float* __restrict__ x,
                     float* __restrict__ out_d,
                     int* __restrict__ out_i,
                     int n)
{
    __shared__ float4 cpts[2][CT];           // double-buffered column points: (x,y,z,|p|^2)
    __shared__ float  dtile[ROWS * LDSW];    // 128x64 distance tile, padded stride

    const int tid  = threadIdx.x;            // 0..127
    const int lane = tid & 31;
    const int wave = tid >> 5;               // 0..3
    const int hi   = lane >> 4;              // lane half: 0 or 1
    const int l16  = lane & 15;
    const int rowbase = blockIdx.x * ROWS;

    // ---- A fragments: 2 row-tiles of 16 rows each per wave, loop-invariant ----
    // A is 16x4 f32: lanes 0-15 hold K=0 (v0) / K=1 (v1); lanes 16-31 hold K=2 / K=3.
    // K payload: (-2x, -2y, -2z, |row|^2)   =>  A.B_col(x,y,z,1) = |row|^2 - 2*dot
    v2f afrag[2];
    #pragma unroll
    for (int t = 0; t < 2; ++t) {
        int r = rowbase + (wave * 2 + t) * 16 + l16;
        float px = x[3 * r + 0], py = x[3 * r + 1], pz = x[3 * r + 2];
        float rs = px * px + py * py + pz * pz;
        afrag[t][0] = hi ? (-2.0f * pz) : (-2.0f * px);
        afrag[t][1] = hi ? rs           : (-2.0f * py);
    }

    // ---- per-thread top-k registers (sorted ascending) ----
    float bd[KSEL];
    int   bi[KSEL];
    #pragma unroll
    for (int i = 0; i < KSEL; ++i) { bd[i] = 3.0e38f; bi[i] = -1; }
    const int myrow = rowbase + tid;

    // ---- prologue: stage chunk 0 ----
    if (tid < CT) {
        float px = x[3 * tid + 0], py = x[3 * tid + 1], pz = x[3 * tid + 2];
        cpts[0][tid] = make_float4(px, py, pz, px * px + py * py + pz * pz);
    }
    __syncthreads();

    const int nchunks = n / CT;
    for (int it = 0; it < nchunks; ++it) {
        const int cbase = it * CT;
        const int cur   = it & 1;

        // ---- issue next-chunk global loads early; latency hidden under WMMA phase ----
        float nx = 0.0f, ny = 0.0f, nz = 0.0f;
        const bool stager = (tid < CT) & (it + 1 < nchunks);
        if (stager) {
            int c = cbase + CT + tid;
            nx = x[3 * c + 0]; ny = x[3 * c + 1]; nz = x[3 * c + 2];
        }

        // ---- WMMA distance tiles: each wave does 2 row-tiles x 4 col-tiles ----
        #pragma unroll
        for (int ct = 0; ct < 4; ++ct) {
            float4 cp = cpts[cur][ct * 16 + l16];
            // B is 4x16 f32: v0 holds K=0 (lanes 0-15) / K=2 (lanes 16-31); v1 = K=1 / K=3
            v2f bfrag;
            bfrag[0] = hi ? cp.z : cp.x;
            bfrag[1] = hi ? 1.0f : cp.y;
            float csq = cp.w;                         // |col|^2, same both halves

            #pragma unroll
            for (int t = 0; t < 2; ++t) {
                v8f acc = {};
                acc = __builtin_amdgcn_wmma_f32_16x16x4_f32(
                        false, afrag[t], false, bfrag,
                        (short)0, acc, false, false);
                // acc[v] = |row|^2 - 2*dot  for row = tilebase + v (+8 for hi half)
                int trow = (wave * 2 + t) * 16 + hi * 8;
                int base = trow * LDSW + ct * 16 + l16;
                #pragma unroll
                for (int v = 0; v < 8; ++v) {
                    float dv = fmaxf(acc[v] + csq, 0.0f);   // clamp like reference
                    dtile[base + v * LDSW] = dv;
                }
            }
        }

        // ---- park next chunk's points into the alternate buffer ----
        // Readers of cpts[!cur] (compute of it-1) finished before that iteration's
        // mid-barrier; visibility to it+1's compute is ordered by this loop's barriers.
        if (stager) {
            cpts[cur ^ 1][tid] = make_float4(nx, ny, nz, nx * nx + ny * ny + nz * nz);
        }
        __syncthreads();

        // ---- scan: each thread owns one row; 4-wide min pre-filter + rare insert ----
        const float* rowp = &dtile[tid * LDSW];       // 272B base: 16B aligned
        for (int j = 0; j < CT; j += 4) {
            float4 d4 = *(const float4*)(rowp + j);   // ds_load_b128
            float m = fminf(fminf(d4.x, d4.y), fminf(d4.z, d4.w));
            if (m < bd[KSEL - 1]) {                   // common path: 3 min + 1 cmp per 4
                float ds[4] = { d4.x, d4.y, d4.z, d4.w };
                #pragma unroll
                for (int u = 0; u < 4; ++u) {
                    float d = ds[u];
                    if (d < bd[KSEL - 1]) {
                        int jg = cbase + j + u;
                        if (jg != myrow) {            // reference SELF_MASK semantics
                            int pos = KSEL - 1;
                            #pragma unroll
                            for (int p = KSEL - 1; p > 0; --p) {
                                if (d < bd[p - 1]) {
                                    bd[p] = bd[p - 1]; bi[p] = bi[p - 1]; pos = p - 1;
                                }
                            }
                            bd[pos] = d;
                            bi[pos] = jg;
                        }
                    }
                }
            }
        }
        __syncthreads();   // scan done before next compute overwrites dtile / reads cpts
    }

    // ---- write results: dists then indices (reference tuple order) ----
    #pragma unroll
    for (int i = 0; i < KSEL; ++i) {
        out_d[(size_t)myrow * KSEL + i] = bd[i];
        out_i[(size_t)myrow * KSEL + i] = bi[i];
    }
}

extern "C" void kernel_launch(void* const* d_in, const int* in_sizes, int n_in,
                              void* d_out, int out_size, void* d_ws, size_t ws_size,
                              hipStream_t stream) {
    const float* x = (const float*)d_in[0];
    int n = in_sizes[0] / 3;                 // 65536
    float* out_d = (float*)d_out;            // (N, 12) float32 distances
    int*   out_i = (int*)((float*)d_out + (size_t)n * KSEL);  // (N, 12) int32 indices

    int blocks = n / ROWS;                   // 512 blocks of 128 threads
    knn_wmma_kernel<<<blocks, ROWS, 0, stream>>>(x, out_d, out_i, n);
}